// lossFunc_33638183862597
// MI455X (gfx1250) — compile-verified
//
#include <hip/hip_runtime.h>

typedef float v2f __attribute__((ext_vector_type(2)));
typedef float v4f __attribute__((ext_vector_type(4)));
typedef float v8f __attribute__((ext_vector_type(8)));
typedef int   v4i __attribute__((ext_vector_type(4)));

#define NBLOCKS 2048
#define NTHREADS 256

// Reduce a per-lane fp32 partial across the wave32 using one
// v_wmma_f32_16x16x4_f32: A = {v, 0} per lane (16x4 matrix holds all 32 lane
// values exactly once), B = all-ones 4x16, C = 0.  D[m][n] = sum_k A[m][k]
// (broadcast across n).  Lanes 0-15 hold rows M=0..7 in D's 8 VGPRs, lanes
// 16-31 hold rows M=8..15, so local sum of the 8 D components + one xor-16
// shuffle yields the full wave sum in every lane.
__device__ __forceinline__ float wave_reduce_wmma(float v) {
    v2f a;
    a[0] = v;
    a[1] = 0.0f;
    v2f b;
    b[0] = 1.0f;
    b[1] = 1.0f;
    v8f c = {};
    v8f d = __builtin_amdgcn_wmma_f32_16x16x4_f32(
        /*neg_a=*/false, a, /*neg_b=*/false, b,
        /*c_mod=*/(short)0, c, /*reuse_a=*/false, /*reuse_b=*/false);
    float half = d[0] + d[1] + d[2] + d[3] + d[4] + d[5] + d[6] + d[7];
    float other = __shfl_xor(half, 16, 32);
    return half + other;
}

__global__ void __launch_bounds__(NTHREADS)
bce_stream_kernel(const float* __restrict__ pred,
                  const float* __restrict__ labels,
                  const int* __restrict__ scores,
                  float* __restrict__ bp,     // d_out + 1          (4B-misaligned for 16B)
                  float* __restrict__ gt,     // d_out + 1 + n
                  float* __restrict__ psum,   // d_ws[0..NBLOCKS)
                  float* __restrict__ pcnt,   // d_ws[NBLOCKS..2*NBLOCKS)
                  long long n) {
    const long long stride = (long long)gridDim.x * blockDim.x * 4;
    long long i = ((long long)blockIdx.x * blockDim.x + threadIdx.x) * 4;

    float s = 0.0f;   // masked bce sum
    float cnt = 0.0f; // mask count

    for (; i < n; i += stride) {
        if (i + 3 < n) {
            // Streaming path: B128 NT loads (inputs are 16B aligned since i%4==0),
            // b32 NT stores (output regions are +4B offset -> not 16B aligned).
            v4f x = __builtin_nontemporal_load((const v4f*)(pred + i));
            v4f y = __builtin_nontemporal_load((const v4f*)(labels + i));
            v4i m = __builtin_nontemporal_load((const v4i*)(scores + i));
#pragma unroll
            for (int k = 0; k < 4; ++k) {
                float xk = x[k], yk = y[k];
                // max(x,0) - x*y + log1p(exp(-|x|))
                float bce = fmaxf(xk, 0.0f) - xk * yk +
                            __logf(1.0f + __expf(-fabsf(xk)));
                bool msk = (m[k] == 1);
                float bv = msk ? xk : 0.0f;
                float gv = msk ? yk : 0.0f;
                s   += msk ? bce : 0.0f;
                cnt += msk ? 1.0f : 0.0f;
                __builtin_nontemporal_store(bv, bp + i + k);
                __builtin_nontemporal_store(gv, gt + i + k);
            }
        } else {
            for (int k = 0; k < 4; ++k) {
                long long j = i + k;
                if (j >= n) break;
                float xk = pred[j], yk = labels[j];
                float bce = fmaxf(xk, 0.0f) - xk * yk +
                            __logf(1.0f + __expf(-fabsf(xk)));
                bool msk = (scores[j] == 1);
                bp[j] = msk ? xk : 0.0f;
                gt[j] = msk ? yk : 0.0f;
                s   += msk ? bce : 0.0f;
                cnt += msk ? 1.0f : 0.0f;
            }
        }
    }

    // EXEC is all-ones here (loop reconverged) -> WMMA wave reductions legal.
    float wsum = wave_reduce_wmma(s);
    float wcnt = wave_reduce_wmma(cnt);

    __shared__ float lsum[NTHREADS / 32];
    __shared__ float lcnt[NTHREADS / 32];
    const int lane = threadIdx.x & 31;
    const int wave = threadIdx.x >> 5;
    if (lane == 0) {
        lsum[wave] = wsum;
        lcnt[wave] = wcnt;
    }
    __syncthreads();
    if (threadIdx.x == 0) {
        float S = 0.0f, C = 0.0f;
#pragma unroll
        for (int w = 0; w < NTHREADS / 32; ++w) {
            S += lsum[w];
            C += lcnt[w];
        }
        // Fixed per-block slots: no atomics, deterministic, no ws init needed.
        psum[blockIdx.x] = S;
        pcnt[blockIdx.x] = C;
    }
}

__global__ void __launch_bounds__(256)
bce_finalize_kernel(const float* __restrict__ psum,
                    const float* __restrict__ pcnt,
                    float* __restrict__ out) {
    __shared__ float ss[256];
    __shared__ float sc[256];
    float s = 0.0f, c = 0.0f;
    for (int i = threadIdx.x; i < NBLOCKS; i += 256) {
        s += psum[i];
        c += pcnt[i];
    }
    ss[threadIdx.x] = s;
    sc[threadIdx.x] = c;
    __syncthreads();
    for (int off = 128; off > 0; off >>= 1) {
        if ((int)threadIdx.x < off) {
            ss[threadIdx.x] += ss[threadIdx.x + off];
            sc[threadIdx.x] += sc[threadIdx.x + off];
        }
        __syncthreads();
    }
    if (threadIdx.x == 0) {
        float C = sc[0];
        out[0] = ss[0] / C / C;
    }
}

extern "C" void kernel_launch(void* const* d_in, const int* in_sizes, int n_in,
                              void* d_out, int out_size, void* d_ws, size_t ws_size,
                              hipStream_t stream) {
    const float* pred   = (const float*)d_in[0];
    const float* labels = (const float*)d_in[1];
    const int*   scores = (const int*)d_in[2];
    const long long n = (long long)in_sizes[0];

    float* out  = (float*)d_out;
    float* bp   = out + 1;            // binary_preds
    float* gt   = out + 1 + n;        // ground_truths
    float* psum = (float*)d_ws;       // per-block masked-bce sums
    float* pcnt = psum + NBLOCKS;     // per-block mask counts

    bce_stream_kernel<<<NBLOCKS, NTHREADS, 0, stream>>>(
        pred, labels, scores, bp, gt, psum, pcnt, n);
    bce_finalize_kernel<<<1, 256, 0, stream>>>(psum, pcnt, out);
}